// OctreeConvBn_8624294331028
// MI455X (gfx1250) — compile-verified
//
#include <hip/hip_runtime.h>

typedef __attribute__((ext_vector_type(2))) float v2f;
typedef __attribute__((ext_vector_type(8))) float v8f;

#define C_IN   16
#define C_OUT  32
#define KNB    27
#define RTOT   (C_IN * KNB)      // 432 reduction length
#define HT     64                // h positions per block
#define COLP   (HT + 1)          // LDS pitch (bank-conflict padding)
#define NCHUNK (RTOT / 4)        // 108 WMMA K-steps (K=4 each)

// ---------------- zero the BN accumulators in d_ws ----------------
__global__ void ocbn_zero_ws(float* __restrict__ ws) {
    if (threadIdx.x < 2 * C_OUT) ws[threadIdx.x] = 0.0f;
}

// ---------------- transpose x(16,H) -> xT(H,16) so each gather is 1 cacheline ----
__global__ __launch_bounds__(256)
void ocbn_transpose(const float* __restrict__ x, float* __restrict__ xT, int H) {
    int t = blockIdx.x * 256 + threadIdx.x;   // 4 lanes per h, each does 4 channels
    int h = t >> 2;
    int q = t & 3;
    if (h >= H) return;
    float4 v;
    v.x = x[(size_t)(4 * q + 0) * H + h];
    v.y = x[(size_t)(4 * q + 1) * H + h];
    v.z = x[(size_t)(4 * q + 2) * H + h];
    v.w = x[(size_t)(4 * q + 3) * H + h];
    ((float4*)xT)[(size_t)h * 4 + q] = v;     // lanes write consecutive float4s
}

// ---------------- gather + fp32 WMMA conv + BN partial sums ----------------
__global__ __launch_bounds__(128)
void ocbn_conv_wmma(const float* __restrict__ x,
                    const float* __restrict__ xT,
                    const float* __restrict__ w,
                    const long long* __restrict__ neigh,
                    float* __restrict__ out,
                    float* __restrict__ ws,
                    int H, int useXT) {
    extern __shared__ float smem[];
    float* col   = smem;                      // [RTOT][COLP]
    float* s_sum = smem + RTOT * COLP;        // [C_OUT]
    float* s_sq  = s_sum + C_OUT;             // [C_OUT]

    const int tid = threadIdx.x;
    const int h0  = blockIdx.x * HT;

    if (tid < C_OUT)            s_sum[tid] = 0.0f;
    else if (tid < 2 * C_OUT)   s_sq[tid - C_OUT] = 0.0f;

    // ---- stage gathered col[r][hl] into LDS (zero-fill pads & invalid idx) ----
    if (useXT) {
        // 4 lanes per (k,hl) pair; each lane loads one float4 (4 channels) of xT[idx]
        for (int t = tid; t < KNB * HT * 4; t += 128) {
            int j    = t & 3;        // channel quarter
            int pair = t >> 2;
            int hl   = pair & (HT - 1);
            int k    = pair >> 6;    // HT == 64
            int h    = h0 + hl;
            float4 v = make_float4(0.f, 0.f, 0.f, 0.f);
            if (h < H) {
                long long ll = neigh[(long long)h * KNB + k];
                if (ll >= 0) v = ((const float4*)xT)[(size_t)ll * 4 + j];
            }
            int rb = (4 * j) * KNB + k;
            col[(rb + 0 * KNB) * COLP + hl] = v.x;
            col[(rb + 1 * KNB) * COLP + hl] = v.y;
            col[(rb + 2 * KNB) * COLP + hl] = v.z;
            col[(rb + 3 * KNB) * COLP + hl] = v.w;
        }
    } else {
        // fallback: strided gather directly from channel-major x
        for (int t = tid; t < KNB * HT; t += 128) {
            int hl = t & (HT - 1);
            int k  = t >> 6;
            int h  = h0 + hl;
            long long idx = -1;
            if (h < H) idx = neigh[(long long)h * KNB + k];
#pragma unroll
            for (int i = 0; i < C_IN; ++i) {
                float v = (idx >= 0) ? x[(size_t)i * H + (size_t)idx] : 0.0f;
                col[(i * KNB + k) * COLP + hl] = v;
            }
        }
    }
    __syncthreads();

    // ---- WMMA: each wave computes a 32(out-ch) x 16(h) tile ----
    const int wid   = tid >> 5;         // wave32
    const int lane  = tid & 31;
    const int n     = lane & 15;        // N (h) / M (out-ch) index within half-wave
    const int halfK = (lane >> 4) * 2;  // lanes 0-15 -> K {0,1}; lanes 16-31 -> K {2,3}
    const int hw    = wid << 4;         // wave's h offset inside block tile

    v8f c0 = {};   // out channels 0..15
    v8f c1 = {};   // out channels 16..31
    const float* w0 = w + (size_t)n * RTOT;          // A rows: M = n
    const float* w1 = w + (size_t)(n + 16) * RTOT;

#pragma unroll 4
    for (int c = 0; c < NCHUNK; ++c) {
        int r0 = 4 * c + halfK;
        v2f a0 = *(const v2f*)(w0 + r0);             // A[M=n][K=r0, r0+1]
        v2f a1 = *(const v2f*)(w1 + r0);
        v2f b;
        b.x = col[(r0 + 0) * COLP + hw + n];         // B[K=r0  ][N=n]
        b.y = col[(r0 + 1) * COLP + hw + n];         // B[K=r0+1][N=n]
        c0 = __builtin_amdgcn_wmma_f32_16x16x4_f32(false, a0, false, b,
                                                   (short)0, c0, false, false);
        c1 = __builtin_amdgcn_wmma_f32_16x16x4_f32(false, a1, false, b,
                                                   (short)0, c1, false, false);
    }

    // ---- store tile + accumulate BN partial sums (padded h contribute 0) ----
    const int hg    = h0 + hw + n;
    const bool inH  = (hg < H);
    const int mbase = (lane < 16) ? 0 : 8;           // C/D layout: VGPR j -> M = j (+8)
#pragma unroll
    for (int j = 0; j < 8; ++j) {
        int ch = mbase + j;
        float y0 = c0[j];
        float y1 = c1[j];
        if (inH) {
            out[(size_t)ch * H + hg]        = y0;
            out[(size_t)(ch + 16) * H + hg] = y1;
        }
        atomicAdd(&s_sum[ch],      y0);
        atomicAdd(&s_sq[ch],       y0 * y0);
        atomicAdd(&s_sum[ch + 16], y1);
        atomicAdd(&s_sq[ch + 16],  y1 * y1);
    }
    __syncthreads();

    if (tid < C_OUT)          unsafeAtomicAdd(&ws[tid], s_sum[tid]);
    else if (tid < 2 * C_OUT) unsafeAtomicAdd(&ws[tid], s_sq[tid - C_OUT]);
}

// ---------------- batch-norm normalization in place ----------------
__global__ __launch_bounds__(256)
void ocbn_bn(float* __restrict__ out, const float* __restrict__ ws,
             const float* __restrict__ gamma, const float* __restrict__ beta, int H) {
    int o = blockIdx.y;
    int h = blockIdx.x * 256 + threadIdx.x;
    if (h >= H) return;
    float inv_h = 1.0f / (float)H;
    float mean  = ws[o] * inv_h;
    float var   = ws[C_OUT + o] * inv_h - mean * mean;
    float s     = rsqrtf(var + 1e-5f) * gamma[o];
    float b     = beta[o] - mean * s;
    size_t i    = (size_t)o * H + h;
    out[i] = out[i] * s + b;
}

extern "C" void kernel_launch(void* const* d_in, const int* in_sizes, int n_in,
                              void* d_out, int out_size, void* d_ws, size_t ws_size,
                              hipStream_t stream) {
    const float*     x     = (const float*)d_in[0];      // (16, H)
    const float*     w     = (const float*)d_in[1];      // (32, 16, 27)
    const float*     gamma = (const float*)d_in[2];
    const float*     beta  = (const float*)d_in[3];
    const long long* neigh = (const long long*)d_in[4];  // (H, 27) int64
    (void)n_in; (void)out_size;

    const int H = in_sizes[0] / C_IN;
    float* out = (float*)d_out;
    float* ws  = (float*)d_ws;

    const size_t xT_bytes = (size_t)H * C_IN * sizeof(float);
    const int useXT = (ws_size >= 256 + xT_bytes) ? 1 : 0;
    float* xT = (float*)((char*)d_ws + 256);

    ocbn_zero_ws<<<1, 64, 0, stream>>>(ws);

    if (useXT) {
        int work = H * 4;
        ocbn_transpose<<<(work + 255) / 256, 256, 0, stream>>>(x, xT, H);
    }

    const int nblk = (H + HT - 1) / HT;
    const size_t shmem = (size_t)(RTOT * COLP + 2 * C_OUT) * sizeof(float);
    hipFuncSetAttribute(reinterpret_cast<const void*>(ocbn_conv_wmma),
                        hipFuncAttributeMaxDynamicSharedMemorySize, (int)shmem);
    ocbn_conv_wmma<<<nblk, 128, shmem, stream>>>(x, xT, w, neigh, out, ws, H, useXT);

    ocbn_bn<<<dim3((H + 255) / 256, C_OUT), 256, 0, stream>>>(out, ws, gamma, beta, H);
}